// GPool_64192581206788
// MI455X (gfx1250) — compile-verified
//
#include <hip/hip_runtime.h>
#include <math.h>

typedef float v2f __attribute__((ext_vector_type(2)));
typedef float v8f __attribute__((ext_vector_type(8)));

#define NB 64
#define CH 128
#define TT 256
#define VV 25
#define KK 3
#define VKEEP 12
#define CT (CH * TT)                 // 32768
#define NCHUNK 8
#define ROWS_PER_CHUNK (CT / NCHUNK) // 4096

// workspace layout (units: floats / ints)
#define WS_SCALE   0
#define WS_PARTIAL 64                               // NB*NCHUNK*VV = 12800 floats
#define WS_IDX     (WS_PARTIAL + NB * NCHUNK * VV)  // NB*VKEEP ints
#define WS_YHAT    (WS_IDX + NB * VKEEP)            // NB*VKEEP floats

// ---------------------------------------------------------------- p-norm ----
__global__ void pnorm_kernel(const float* __restrict__ p, float* __restrict__ ws) {
    __shared__ float red[1024];
    int tid = threadIdx.x;
    float s = 0.f;
    for (int i = tid; i < CT; i += 1024) { float v = p[i]; s += v * v; }
    red[tid] = s;
    __syncthreads();
    for (int st = 512; st > 0; st >>= 1) {
        if (tid < st) red[tid] += red[tid + st];
        __syncthreads();
    }
    if (tid == 0) ws[WS_SCALE] = rsqrtf(red[0]);
}

// ----------------------------------------------- scores: partial reduction --
// grid (NCHUNK, NB) x 256 threads. Deterministic LDS tree reduce (no atomics).
__global__ void score_partial_kernel(const float* __restrict__ x,
                                     const float* __restrict__ p,
                                     float* __restrict__ ws) {
    int chunk = blockIdx.x, n = blockIdx.y, tid = threadIdx.x;
    float acc[VV];
#pragma unroll
    for (int v = 0; v < VV; ++v) acc[v] = 0.f;

    const float* xb = x + (size_t)n * CT * VV;
    int row0 = chunk * ROWS_PER_CHUNK;
    for (int r = row0 + tid; r < row0 + ROWS_PER_CHUNK; r += 256) {
        float pv = p[r];
        const float* xr = xb + (size_t)r * VV;
#pragma unroll
        for (int v = 0; v < VV; ++v) acc[v] += pv * xr[v];
    }

    __shared__ float red[256 * VV];
#pragma unroll
    for (int v = 0; v < VV; ++v) red[tid * VV + v] = acc[v];
    __syncthreads();
    for (int s = 128; s > 0; s >>= 1) {
        if (tid < s) {
#pragma unroll
            for (int v = 0; v < VV; ++v) red[tid * VV + v] += red[(tid + s) * VV + v];
        }
        __syncthreads();
    }
    if (tid < VV) ws[WS_PARTIAL + (n * NCHUNK + chunk) * VV + tid] = red[tid];
}

// ----------------------------------------------------------- top-k select --
// one thread per batch; strict '>' => smallest index wins ties (stable argsort)
__global__ void topk_kernel(float* __restrict__ ws) {
    int n = threadIdx.x;
    if (n >= NB) return;
    float scale = ws[WS_SCALE];
    float ys[VV];
#pragma unroll
    for (int v = 0; v < VV; ++v) {
        float s = 0.f;
        for (int c = 0; c < NCHUNK; ++c) s += ws[WS_PARTIAL + (n * NCHUNK + c) * VV + v];
        ys[v] = s * scale;
    }
    int* idx = (int*)ws + WS_IDX;
    bool used[VV];
#pragma unroll
    for (int v = 0; v < VV; ++v) used[v] = false;
    for (int j = 0; j < VKEEP; ++j) {
        int best = 0;
        float bv = -INFINITY;
#pragma unroll
        for (int v = 0; v < VV; ++v) {
            if (!used[v] && ys[v] > bv) { bv = ys[v]; best = v; }
        }
        used[best] = true;
        idx[n * VKEEP + j] = best;
        ws[WS_YHAT + n * VKEEP + j] = 1.f / (1.f + expf(-bv));
    }
}

// ------------------------------------------------------- x gather + scale --
// one thread per output element; CT*VKEEP = 393216 is a multiple of 256, so
// every block lives inside a single n -> stage idx/y_hat in LDS once.
__global__ void gather_x_kernel(const float* __restrict__ x,
                                const float* __restrict__ ws,
                                float* __restrict__ out) {
    __shared__ int   sidx[VKEEP];
    __shared__ float syh[VKEEP];
    int tid = threadIdx.x;
    long long o = (long long)blockIdx.x * 256 + tid;
    int n = (int)(o / ((long long)CT * VKEEP));
    if (tid < VKEEP) {
        sidx[tid] = ((const int*)ws)[WS_IDX + n * VKEEP + tid];
        syh[tid]  = ws[WS_YHAT + n * VKEEP + tid];
    }
    __syncthreads();
    int rem = (int)(o - (long long)n * CT * VKEEP);
    int ct = rem / VKEEP;
    int j  = rem - ct * VKEEP;
    out[o] = x[((size_t)n * CT + ct) * VV + sidx[j]] * syh[j];
}

// -------------------------------------------- A2 = A@A via WMMA + gather ----
// one wave32 per (n,k): pad 25x25 -> 32x32 in LDS, 2x2 tiles of 16x16, K in
// 8 steps of 4 using V_WMMA_F32_16X16X4_F32, then gather A2[idx[i], idx[j]].
__global__ void __launch_bounds__(32)
a2_wmma_gather_kernel(const float* __restrict__ A,
                      const float* __restrict__ ws,
                      float* __restrict__ aout) {
    __shared__ float Ap[32 * 32];
    __shared__ float A2[32 * 32];
    int l = threadIdx.x;   // 0..31, EXEC all-ones (no divergence before wmma)
    int m = blockIdx.x;    // n*KK + k
    const float* Ain = A + (size_t)m * (VV * VV);

    for (int t = l; t < 1024; t += 32) {            // uniform 32 iters per lane
        int r = t >> 5, c = t & 31;
        Ap[t] = (r < VV && c < VV) ? Ain[r * VV + c] : 0.f;
    }
    __syncthreads();

    int lh   = l & 15;            // M / N index within tile
    int koff = (l >> 4) << 1;     // lanes 16-31 hold K+2, K+3 (ISA 7.12.2)
    v8f c00 = {}, c01 = {}, c10 = {}, c11 = {};

#pragma unroll
    for (int k0 = 0; k0 < 32; k0 += 4) {
        v2f a0, a1, b0, b1;
        // A-fragment 16x4: lane lh = row M, vgpr j = K (k0 + j + koff)
        a0.x = Ap[lh * 32 + k0 + koff];
        a0.y = Ap[lh * 32 + k0 + 1 + koff];
        a1.x = Ap[(lh + 16) * 32 + k0 + koff];
        a1.y = Ap[(lh + 16) * 32 + k0 + 1 + koff];
        // B-fragment 4x16: lane lh = col N, vgpr j = K row
        b0.x = Ap[(k0 + koff) * 32 + lh];
        b0.y = Ap[(k0 + 1 + koff) * 32 + lh];
        b1.x = Ap[(k0 + koff) * 32 + 16 + lh];
        b1.y = Ap[(k0 + 1 + koff) * 32 + 16 + lh];

        c00 = __builtin_amdgcn_wmma_f32_16x16x4_f32(false, a0, false, b0, (short)0, c00, false, false);
        c01 = __builtin_amdgcn_wmma_f32_16x16x4_f32(false, a0, false, b1, (short)0, c01, false, false);
        c10 = __builtin_amdgcn_wmma_f32_16x16x4_f32(false, a1, false, b0, (short)0, c10, false, false);
        c11 = __builtin_amdgcn_wmma_f32_16x16x4_f32(false, a1, false, b1, (short)0, c11, false, false);
    }

    // C/D layout: VGPR r holds rows r (lanes 0-15) and r+8 (lanes 16-31)
    int rowoff = (l >> 4) << 3;   // 0 or 8
#pragma unroll
    for (int r = 0; r < 8; ++r) {
        int row = r + rowoff;
        A2[row * 32 + lh]             = c00[r];
        A2[row * 32 + 16 + lh]        = c01[r];
        A2[(row + 16) * 32 + lh]      = c10[r];
        A2[(row + 16) * 32 + 16 + lh] = c11[r];
    }
    __syncthreads();

    int n = m / KK;
    const int* idxn = (const int*)ws + WS_IDX + n * VKEEP;
    for (int t = l; t < VKEEP * VKEEP; t += 32) {
        int i = t / VKEEP, j = t - i * VKEEP;
        aout[(size_t)m * (VKEEP * VKEEP) + t] = A2[idxn[i] * 32 + idxn[j]];
    }
}

// ---------------------------------------------------------------------------
extern "C" void kernel_launch(void* const* d_in, const int* in_sizes, int n_in,
                              void* d_out, int out_size, void* d_ws, size_t ws_size,
                              hipStream_t stream) {
    const float* x = (const float*)d_in[0];   // [N, C, T, V]
    const float* A = (const float*)d_in[1];   // [N, K, V, V]
    const float* p = (const float*)d_in[2];   // [C*T, 1]
    float* ws   = (float*)d_ws;
    float* xout = (float*)d_out;                               // [N, C, T, Vk]
    float* aout = xout + (size_t)NB * CH * TT * VKEEP;         // [N, K, Vk, Vk]

    pnorm_kernel<<<1, 1024, 0, stream>>>(p, ws);
    score_partial_kernel<<<dim3(NCHUNK, NB), 256, 0, stream>>>(x, p, ws);
    topk_kernel<<<1, 64, 0, stream>>>(ws);
    gather_x_kernel<<<(NB * (size_t)CT * VKEEP) / 256, 256, 0, stream>>>(x, ws, xout);
    a2_wmma_gather_kernel<<<NB * KK, 32, 0, stream>>>(A, ws, aout);
}